// LCL_16879221473598
// MI455X (gfx1250) — compile-verified
//
#include <hip/hip_runtime.h>
#include <stdint.h>

// Problem geometry (fixed by the reference).
#define BATCH  32
#define H      1024
#define W      1024
#define NPIX   (H * W)               // 1<<20 per batch
#define TOTALF 33554432.0f           // 32 * 1<<20 (2^25, exact in fp32)
#define K_RANK 838860u               // floor(0.8*(NPIX-1)); fp32 frac == 0 -> exact order stat
#define NBIN   2048
#define TR     8                     // rows computed per tile
#define TLR    (TR + 2)              // + top/bottom halo
#define TILES_PER_B (H / TR)         // 128
#define NWG    (BATCH * TILES_PER_B) // 4096 workgroups for stencil passes
#define EPS    1e-6f

// ---- gfx1250 async global->LDS staging (ASYNCcnt path), guarded ----------
#if defined(__HIP_DEVICE_COMPILE__) && __has_builtin(__builtin_amdgcn_global_load_async_to_lds_b128)
#define USE_ASYNC_LDS 1
#else
#define USE_ASYNC_LDS 0
#endif

typedef int lcl_v4i __attribute__((vector_size(16)));
#if USE_ASYNC_LDS
typedef __attribute__((address_space(1))) lcl_v4i g_v4i;   // global (AS1) int4
typedef __attribute__((address_space(3))) lcl_v4i l_v4i;   // LDS (AS3) int4
#endif

__device__ __forceinline__ void cp_g2l_b128(float* lds_dst, const float* gsrc) {
#if USE_ASYNC_LDS
    // AS1 pointer value == flat pointer value for global memory, so the
    // integer round-trip is value-preserving. LDS needs a genuine
    // addrspacecast (C-style cast emits it and strips the aperture).
    __builtin_amdgcn_global_load_async_to_lds_b128(
        (g_v4i*)(uintptr_t)gsrc,
        (l_v4i*)lds_dst,
        0, 0);
#else
    *(float4*)lds_dst = *(const float4*)gsrc;
#endif
}

__device__ __forceinline__ void cp_g2l_wait() {
#if USE_ASYNC_LDS
#if __has_builtin(__builtin_amdgcn_s_wait_asynccnt)
    __builtin_amdgcn_s_wait_asynccnt(0);
#else
    asm volatile("s_wait_asynccnt 0" ::: "memory");
#endif
#endif
}

// Issue tile loads: TLR rows x 1024 cols into LDS, zero-filled halo at image edges.
// Thread t owns float4 chunk t of every row (256 chunks == blockDim).
__device__ __forceinline__ void tile_issue(float* tile, const float* base, int r0, int t) {
#pragma unroll
    for (int lr = 0; lr < TLR; ++lr) {
        int gr = r0 - 1 + lr;
        float* dst = tile + lr * W + t * 4;
        if (gr < 0 || gr >= H) {
            *(float4*)dst = make_float4(0.f, 0.f, 0.f, 0.f);
        } else {
            cp_g2l_b128(dst, base + (size_t)gr * W + t * 4);
        }
    }
}

// 5-point Laplacian magnitude for 4 consecutive pixels, entirely from LDS.
__device__ __forceinline__ void lap4(const float* tile, int rr, int c4, float L[4]) {
    const float* row  = tile + (rr + 1) * W;
    const float* rowu = row - W;
    const float* rowd = row + W;
    float4 m = *(const float4*)(row + c4);
    float4 u = *(const float4*)(rowu + c4);
    float4 d = *(const float4*)(rowd + c4);
    float lf = (c4 == 0)     ? 0.f : row[c4 - 1];
    float rt = (c4 == W - 4) ? 0.f : row[c4 + 4];
    L[0] = fabsf(u.x + d.x + lf  + m.y - 4.f * m.x);
    L[1] = fabsf(u.y + d.y + m.x + m.z - 4.f * m.y);
    L[2] = fabsf(u.z + d.z + m.y + m.w - 4.f * m.z);
    L[3] = fabsf(u.w + d.w + m.z + rt  - 4.f * m.w);
}

// ---- Pass A (x3 levels): per-batch histogram of L's fp32 bit pattern ------
// level 0: bins = bits[31:21]; level 1: bins = bits[20:10] among matches of
// the selected 11-bit prefix; level 2: bins = bits[9:0] among 22-bit matches.
__global__ __launch_bounds__(256) void k_hist(const float* __restrict__ pred,
                                              unsigned int* __restrict__ ghist,
                                              const unsigned int* __restrict__ prefix,
                                              int level) {
    __shared__ __align__(16) float tile[TLR * W];
    __shared__ unsigned int shist[NBIN];
    const int t  = threadIdx.x;
    const int b  = blockIdx.y;
    const int r0 = blockIdx.x * TR;
    const float* base = pred + (size_t)b * NPIX;

    tile_issue(tile, base, r0, t);            // async copies in flight...
    for (int j = t; j < NBIN; j += 256) shist[j] = 0u;  // ...overlap hist zeroing
    cp_g2l_wait();
    __syncthreads();

    const unsigned int pref = (level > 0) ? prefix[b] : 0u;
#pragma unroll
    for (int rr = 0; rr < TR; ++rr) {
        float L[4];
        lap4(tile, rr, t * 4, L);
#pragma unroll
        for (int j = 0; j < 4; ++j) {
            unsigned int bits = __float_as_uint(L[j]);   // L >= 0 -> order-preserving
            if (level == 0) {
                atomicAdd(&shist[bits >> 21], 1u);
            } else if (level == 1) {
                if ((bits >> 21) == pref) atomicAdd(&shist[(bits >> 10) & 2047u], 1u);
            } else {
                if ((bits >> 10) == pref) atomicAdd(&shist[bits & 1023u], 1u);
            }
        }
    }
    __syncthreads();
    for (int j = t; j < NBIN; j += 256) {
        unsigned int v = shist[j];
        if (v) atomicAdd(&ghist[b * NBIN + j], v);
    }
}

// ---- Pass B (x3 levels): per-batch scan of 2048 bins -> pick bin with rank k
__global__ __launch_bounds__(256) void k_select(const unsigned int* __restrict__ ghist,
                                                unsigned int* __restrict__ prefix,
                                                unsigned int* __restrict__ rank,
                                                float* __restrict__ thresh,
                                                int level) {
    const int b = blockIdx.x, t = threadIdx.x;
    __shared__ unsigned int ssum[256];
    const unsigned int* h = ghist + b * NBIN;

    unsigned int loc[8];
    unsigned int s = 0;
#pragma unroll
    for (int j = 0; j < 8; ++j) { loc[j] = h[t * 8 + j]; s += loc[j]; }

    const unsigned int kk   = (level == 0) ? K_RANK : rank[b];
    const unsigned int pref = (level == 0) ? 0u : prefix[b];

    ssum[t] = s;
    __syncthreads();
    for (int off = 1; off < 256; off <<= 1) {   // Hillis-Steele inclusive scan
        unsigned int v = (t >= off) ? ssum[t - off] : 0u;
        __syncthreads();
        ssum[t] += v;
        __syncthreads();
    }
    const unsigned int incl = ssum[t];
    const unsigned int excl = incl - s;
    __syncthreads();                             // all reads of rank/prefix done

    if (kk >= excl && kk < incl) {               // exactly one owner thread
        unsigned int cum = excl;
        int sel = 0;
        for (int j = 0; j < 8; ++j) {
            if (kk < cum + loc[j]) { sel = t * 8 + j; break; }
            cum += loc[j];
        }
        if (level < 2) {
            prefix[b] = (pref << 11) | (unsigned int)sel;
            rank[b]   = kk - cum;
        } else {
            thresh[b] = __uint_as_float((pref << 10) | (unsigned int)sel);  // exact k-th L
        }
    }
}

// ---- Pass C: masked sums (total sum, edge sum, edge count) per workgroup ---
__global__ __launch_bounds__(256) void k_reduce(const float* __restrict__ pred,
                                                const float* __restrict__ thresh,
                                                float* __restrict__ psa,
                                                float* __restrict__ pse,
                                                unsigned int* __restrict__ pc) {
    __shared__ __align__(16) float tile[TLR * W];
    __shared__ float ra[256];
    __shared__ float re[256];
    __shared__ unsigned int rc[256];
    const int t  = threadIdx.x;
    const int b  = blockIdx.y;
    const int r0 = blockIdx.x * TR;
    const float* base = pred + (size_t)b * NPIX;

    tile_issue(tile, base, r0, t);
    cp_g2l_wait();
    __syncthreads();

    const float th = thresh[b];
    float sa = 0.f, se = 0.f;
    unsigned int c = 0;
#pragma unroll
    for (int rr = 0; rr < TR; ++rr) {
        float L[4];
        lap4(tile, rr, t * 4, L);
#pragma unroll
        for (int j = 0; j < 4; ++j) {
            sa += L[j];
            if (L[j] > th) { se += L[j]; ++c; }
        }
    }
    ra[t] = sa; re[t] = se; rc[t] = c;
    __syncthreads();
    for (int s2 = 128; s2 > 0; s2 >>= 1) {
        if (t < s2) { ra[t] += ra[t + s2]; re[t] += re[t + s2]; rc[t] += rc[t + s2]; }
        __syncthreads();
    }
    if (t == 0) {
        int wg = blockIdx.y * gridDim.x + blockIdx.x;
        psa[wg] = ra[0]; pse[wg] = re[0]; pc[wg] = rc[0];
    }
}

// ---- Pass D: deterministic final reduction + scalar formula ----------------
__global__ __launch_bounds__(256) void k_final(const float* __restrict__ psa,
                                               const float* __restrict__ pse,
                                               const unsigned int* __restrict__ pc,
                                               float* __restrict__ out) {
    __shared__ float ra[256];
    __shared__ float re[256];
    __shared__ unsigned int rc[256];
    const int t = threadIdx.x;
    float sa = 0.f, se = 0.f;
    unsigned int c = 0;
    for (int j = t; j < NWG; j += 256) { sa += psa[j]; se += pse[j]; c += pc[j]; }
    ra[t] = sa; re[t] = se; rc[t] = c;
    __syncthreads();
    for (int s = 128; s > 0; s >>= 1) {
        if (t < s) { ra[t] += ra[t + s]; re[t] += re[t + s]; rc[t] += rc[t + s]; }
        __syncthreads();
    }
    if (t == 0) {
        float edge_sum = re[0];
        float total    = ra[0];
        float edge_cnt = (float)rc[0];            // edge count ~6.7M, exact in fp32
        float flat_sum = total - edge_sum;
        float flat_cnt = TOTALF - edge_cnt;
        float edge_mean = (edge_cnt > 0.f) ? edge_sum / fmaxf(edge_cnt, 1.f) : 0.f;
        float flat_mean = (flat_cnt > 0.f) ? flat_sum / fmaxf(flat_cnt, 1.f) : EPS;
        out[0] = flat_mean / (edge_mean + EPS);
    }
}

extern "C" void kernel_launch(void* const* d_in, const int* in_sizes, int n_in,
                              void* d_out, int out_size, void* d_ws, size_t ws_size,
                              hipStream_t stream) {
    const float* pred = (const float*)d_in[0];

    // Workspace layout (~312 KB total).
    unsigned int* hist   = (unsigned int*)d_ws;          // 32*2048 u32
    unsigned int* prefix = hist + BATCH * NBIN;          // 32 u32
    unsigned int* rank   = prefix + BATCH;               // 32 u32
    float*        thresh = (float*)(rank + BATCH);       // 32 f32
    float*        psa    = thresh + BATCH;               // 4096 f32
    float*        pse    = psa + NWG;                    // 4096 f32
    unsigned int* pc     = (unsigned int*)(pse + NWG);   // 4096 u32

    dim3 grid(TILES_PER_B, BATCH), blk(256);

    // 3-level radix-select on the fp32 bit pattern -> exact per-batch quantile.
    for (int level = 0; level < 3; ++level) {
        (void)hipMemsetAsync(hist, 0, BATCH * NBIN * sizeof(unsigned int), stream);
        k_hist<<<grid, blk, 0, stream>>>(pred, hist, prefix, level);
        k_select<<<BATCH, blk, 0, stream>>>(hist, prefix, rank, thresh, level);
    }
    k_reduce<<<grid, blk, 0, stream>>>(pred, thresh, psa, pse, pc);
    k_final<<<1, blk, 0, stream>>>(psa, pse, pc, (float*)d_out);
}